// BilinearResNet_61976378081964
// MI455X (gfx1250) — compile-verified
//
#include <hip/hip_runtime.h>

typedef __attribute__((ext_vector_type(2))) float v2f;
typedef __attribute__((ext_vector_type(8))) float v8f;

#define BATCH     65536
#define IN_DIM    784
#define D_MODEL   16
#define HIDDEN    6
#define N_BLOCKS  4
#define N_CLASSES 10

#define WAVES   8      // waves per workgroup (wave32)
#define TILE_M  16     // batch rows per wave
#define KC      56     // K-chunk of x staged in LDS (784 = 14 * 56)
#define NCHUNK  (IN_DIM / KC)
#define XS      60     // LDS row stride for x tile   (4*odd -> conflict-free)
#define WES     788    // LDS row stride for W_embed  (788 % 64 = 20 = 4*odd)
#define HXS     20     // LDS row stride for h_x stage
#define HS      12     // LDS row stride for h stage

// dynamic LDS budget (floats)
#define XBUF_F  (TILE_M * XS)                     // 960 per wave per buffer
#define SWE_F   (D_MODEL * WES)                   // 12608
#define SX_F    (3 * WAVES * XBUF_F)              // 23040 (triple buffered)
#define SHX_F   (WAVES * TILE_M * HXS)            // 2560
#define SH_F    (WAVES * TILE_M * HS)             // 1536
#define SMEM_F  (SWE_F + SX_F + SHX_F + SH_F)     // 39744 floats = 158976 B

// ---- CDNA5 async global->LDS copy (ASYNCcnt-tracked, no data VGPRs) ----
__device__ __forceinline__ unsigned lds_addr(const void* p) {
    // low 32 bits of a generic shared-aperture pointer == LDS byte offset
    return (unsigned)(unsigned long long)p;
}
__device__ __forceinline__ void async_ld_b128(unsigned loff, const void* g) {
    unsigned long long ga = (unsigned long long)g;
    asm volatile("global_load_async_to_lds_b128 %0, %1, off"
                 :: "v"(loff), "v"(ga) : "memory");
}
#define WAIT_ASYNC(n) asm volatile("s_wait_asynccnt %0" :: "i"(n) : "memory")

__global__ __launch_bounds__(256) void bilinear_resnet_wmma(
    const float* __restrict__ x,  const float* __restrict__ We,
    const float* __restrict__ Lw, const float* __restrict__ Rw,
    const float* __restrict__ Dw, const float* __restrict__ Wh,
    float* __restrict__ out)
{
    extern __shared__ float smem[];
    float* sWe  = smem;                 // [16][WES]  W_embed, padded
    float* sXb  = smem + SWE_F;         // [WAVES][3][16][XS] x tile chunk ring
    float* sHxb = sXb  + SX_F;          // [WAVES][16][HXS] h_x stage
    float* sHb  = sHxb + SHX_F;         // [WAVES][16][HS]  h stage (cols 0..7)

    const int tid  = threadIdx.x;
    const int wave = tid >> 5;
    const int lane = tid & 31;
    const int l    = lane & 15;   // row (A) / col (B,D) index inside fragment
    const int hf   = lane >> 4;   // half-wave select (K pair / M half)

    // ---- W_embed -> padded LDS via async DMA (coalesced b128, no waits) ----
    for (int t = tid; t < (D_MODEL * IN_DIM) / 4; t += 256) {
        int n  = t / (IN_DIM / 4);          // /196
        int c4 = t - n * (IN_DIM / 4);
        async_ld_b128(lds_addr(sWe + n * WES + c4 * 4),
                      We + n * IN_DIM + c4 * 4);
    }
    WAIT_ASYNC(0);
    __syncthreads();

    float* xwave = sXb + wave * (3 * XBUF_F);   // single LDS base for the ring
    float* hxs   = sHxb + wave * (TILE_M * HXS);
    float* hs    = sHb  + wave * (TILE_M * HS);

    const long rowBase = (long)(blockIdx.x * WAVES + wave) * TILE_M;
    const float* xt = x + rowBase * IN_DIM;

    // per-lane slot geometry for one chunk (7 x b128 per lane)
    int rr[7], cc[7];
#pragma unroll
    for (int it = 0; it < 7; ++it) {
        int t = it * 32 + lane;
        rr[it] = t / (KC / 4);
        cc[it] = (t - rr[it] * (KC / 4)) * 4;
    }

    v8f acc = {};   // h_x tile in C/D layout: row = g + hf*8, col = l

    // ===== embed GEMM: h_x = x_tile @ We^T =====
    // distance-2 async DMA pipeline over a triple-buffered LDS chunk ring;
    // buffer selection by rotating scalar offsets (keeps reads on the DS path).
    {
        unsigned o0 = 0, o1 = XBUF_F, o2 = 2 * XBUF_F;
#pragma unroll
        for (int it = 0; it < 7; ++it)   // issue chunk 0 -> ring slot 0
            async_ld_b128(lds_addr(xwave + o0 + rr[it] * XS + cc[it]),
                          xt + (long)rr[it] * IN_DIM + cc[it]);
#pragma unroll
        for (int it = 0; it < 7; ++it)   // issue chunk 1 -> ring slot 1
            async_ld_b128(lds_addr(xwave + o1 + rr[it] * XS + cc[it]),
                          xt + (long)rr[it] * IN_DIM + KC + cc[it]);

        for (int c = 0; c < NCHUNK; ++c) {
            if (c + 2 < NCHUNK) {        // issue chunk c+2, keep 2 in flight
                const int k0n = (c + 2) * KC;
#pragma unroll
                for (int it = 0; it < 7; ++it)
                    async_ld_b128(lds_addr(xwave + o2 + rr[it] * XS + cc[it]),
                                  xt + (long)rr[it] * IN_DIM + k0n + cc[it]);
                WAIT_ASYNC(14);          // in-order: chunk c has fully landed
            } else if (c + 1 < NCHUNK) {
                WAIT_ASYNC(7);
            } else {
                WAIT_ASYNC(0);
            }
            const float* xp  = xwave + o0 + l * XS + hf * 2;
            const float* wep = sWe + l * WES + c * KC + hf * 2;
            // register-pipelined fragments: load step kk+4 before WMMA(kk)
            v2f a = *(const v2f*)(xp);
            v2f b = *(const v2f*)(wep);
#pragma unroll
            for (int kk = 0; kk < KC; kk += 4) {
                v2f an = a, bn = b;
                if (kk + 4 < KC) {
                    an = *(const v2f*)(xp  + kk + 4);
                    bn = *(const v2f*)(wep + kk + 4);
                }
                acc = __builtin_amdgcn_wmma_f32_16x16x4_f32(
                          false, a, false, b, (short)0, acc, false, false);
                a = an; b = bn;
            }
            unsigned tmp = o0; o0 = o1; o1 = o2; o2 = tmp;   // rotate ring
        }
    }

    const v2f z2 = {0.f, 0.f};

    // ===== bilinear residual blocks =====
#pragma unroll
    for (int i = 0; i < N_BLOCKS; ++i) {
        // stage h_x (C/D layout) -> row-major LDS for use as A
#pragma unroll
        for (int g = 0; g < 8; ++g)
            hxs[(hf * 8 + g) * HXS + l] = acc[g];

        v8f u = {}, v = {};
#pragma unroll
        for (int kk = 0; kk < D_MODEL; kk += 4) {
            v2f a  = *(const v2f*)(hxs + l * HXS + kk + hf * 2);
            v2f bl = (l < HIDDEN)
                   ? *(const v2f*)(Lw + i * HIDDEN * D_MODEL + l * D_MODEL + kk + hf * 2)
                   : z2;
            v2f br = (l < HIDDEN)
                   ? *(const v2f*)(Rw + i * HIDDEN * D_MODEL + l * D_MODEL + kk + hf * 2)
                   : z2;
            u = __builtin_amdgcn_wmma_f32_16x16x4_f32(false, a, false, bl, (short)0, u, false, false);
            v = __builtin_amdgcn_wmma_f32_16x16x4_f32(false, a, false, br, (short)0, v, false, false);
        }
        v8f h = u * v;   // cols >= HIDDEN are exactly zero (zero-padded B)

        // stage h cols 0..7 (6,7 are zeros) for the residual GEMM's A
        if (l < 8) {
#pragma unroll
            for (int g = 0; g < 8; ++g)
                hs[(hf * 8 + g) * HS + l] = h[g];
        }
        // emit h_i to output
        if (l < HIDDEN) {
            float* ho = out + (long)BATCH * N_CLASSES + (long)i * BATCH * HIDDEN;
#pragma unroll
            for (int g = 0; g < 8; ++g)
                ho[(rowBase + hf * 8 + g) * HIDDEN + l] = h[g];
        }
        // residual: acc += h @ Dw[i]^T  (K = 6 padded to 8; padded A cols are 0)
#pragma unroll
        for (int kk = 0; kk < 8; kk += 4) {
            int k = kk + hf * 2;
            v2f a  = *(const v2f*)(hs + l * HS + k);
            v2f bd = (k < HIDDEN)
                   ? *(const v2f*)(Dw + i * D_MODEL * HIDDEN + l * HIDDEN + k)
                   : z2;
            acc = __builtin_amdgcn_wmma_f32_16x16x4_f32(false, a, false, bd, (short)0, acc, false, false);
        }
    }

    // ===== head: logits = h_x @ Wh^T =====
#pragma unroll
    for (int g = 0; g < 8; ++g)
        hxs[(hf * 8 + g) * HXS + l] = acc[g];

    v8f lg = {};
#pragma unroll
    for (int kk = 0; kk < D_MODEL; kk += 4) {
        v2f a  = *(const v2f*)(hxs + l * HXS + kk + hf * 2);
        v2f bh = (l < N_CLASSES)
               ? *(const v2f*)(Wh + l * D_MODEL + kk + hf * 2)
               : z2;
        lg = __builtin_amdgcn_wmma_f32_16x16x4_f32(false, a, false, bh, (short)0, lg, false, false);
    }
    if (l < N_CLASSES) {
#pragma unroll
        for (int g = 0; g < 8; ++g)
            out[(rowBase + hf * 8 + g) * N_CLASSES + l] = lg[g];
    }
}

extern "C" void kernel_launch(void* const* d_in, const int* in_sizes, int n_in,
                              void* d_out, int out_size, void* d_ws, size_t ws_size,
                              hipStream_t stream) {
    const float* x  = (const float*)d_in[0];
    const float* We = (const float*)d_in[1];
    const float* Lw = (const float*)d_in[2];
    const float* Rw = (const float*)d_in[3];
    const float* Dw = (const float*)d_in[4];
    const float* Wh = (const float*)d_in[5];
    float* out = (float*)d_out;

    dim3 grid(BATCH / (WAVES * TILE_M));   // 512 workgroups
    dim3 block(256);                       // 8 waves of 32
    size_t shmem = (size_t)SMEM_F * sizeof(float);   // 158976 B dynamic LDS
    bilinear_resnet_wmma<<<grid, block, shmem, stream>>>(x, We, Lw, Rw, Dw, Wh, out);
}